// GCNet_40733469835916
// MI455X (gfx1250) — compile-verified
//
#include <hip/hip_runtime.h>
#include <hip/hip_bf16.h>

typedef __attribute__((ext_vector_type(2))) float v2f;
typedef __attribute__((ext_vector_type(8))) float v8f;

#define B_   32
#define C_   512
#define C16_ 32
#define HW_  4096
#define LN_EPS_ 1e-5f

// ---------------------------------------------------------------------------
// Pass 1: attention logits  k_raw[b][n] = sum_c x[b][c][n] * w1[c] + b1
// grid: B * (HW/256) = 512 blocks, 256 threads; coalesced over n per c-step.
// ---------------------------------------------------------------------------
__global__ __launch_bounds__(256) void gc_logits(const float* __restrict__ x,
                                                 const float* __restrict__ w1,
                                                 const float* __restrict__ b1,
                                                 float* __restrict__ kbuf) {
    const int b = blockIdx.x >> 4;          // 16 blocks per batch
    const int n = ((blockIdx.x & 15) << 8) + threadIdx.x;
    const float* xb = x + (size_t)b * C_ * HW_;
    float acc = 0.0f;
    #pragma unroll 4
    for (int c = 0; c < C_; ++c)
        acc = fmaf(xb[c * HW_ + n], w1[c], acc);   // w1[c] is a uniform s_load
    kbuf[b * HW_ + n] = acc + b1[0];
}

// ---------------------------------------------------------------------------
// Pass 2: in-place softmax over HW per batch. 32 blocks x 256 threads.
// ---------------------------------------------------------------------------
__global__ __launch_bounds__(256) void gc_softmax(float* __restrict__ kbuf) {
    __shared__ float smem[256];
    float* kb = kbuf + blockIdx.x * HW_;
    const int t = threadIdx.x;

    float m = -INFINITY;
    for (int i = t; i < HW_; i += 256) m = fmaxf(m, kb[i]);
    smem[t] = m; __syncthreads();
    for (int s = 128; s > 0; s >>= 1) {
        if (t < s) smem[t] = fmaxf(smem[t], smem[t + s]);
        __syncthreads();
    }
    m = smem[0]; __syncthreads();

    float sum = 0.0f;
    for (int i = t; i < HW_; i += 256) {
        float e = __expf(kb[i] - m);
        kb[i] = e;
        sum += e;
    }
    smem[t] = sum; __syncthreads();
    for (int s = 128; s > 0; s >>= 1) {
        if (t < s) smem[t] += smem[t + s];
        __syncthreads();
    }
    const float inv = 1.0f / smem[0];
    for (int i = t; i < HW_; i += 256) kb[i] *= inv;
}

// ---------------------------------------------------------------------------
// Pass 3: pooled[b][c] = sum_n x[b][c][n] * k[b][n]
// One 128-thread block per (b,c) row; float4 loads; LDS tree reduction.
// grid: B*C = 16384 blocks.
// ---------------------------------------------------------------------------
__global__ __launch_bounds__(128) void gc_pool(const float* __restrict__ x,
                                               const float* __restrict__ kbuf,
                                               float* __restrict__ pooled) {
    const int b = blockIdx.x >> 9;
    const int c = blockIdx.x & (C_ - 1);
    const float4* xr = (const float4*)(x + ((size_t)b * C_ + c) * HW_);
    const float4* kr = (const float4*)(kbuf + b * HW_);
    float acc = 0.0f;
    #pragma unroll 2
    for (int i = threadIdx.x; i < HW_ / 4; i += 128) {
        float4 xv = xr[i];
        float4 kv = kr[i];
        acc += xv.x * kv.x + xv.y * kv.y + xv.z * kv.z + xv.w * kv.w;
    }
    __shared__ float smem[128];
    smem[threadIdx.x] = acc; __syncthreads();
    for (int s = 64; s > 0; s >>= 1) {
        if (threadIdx.x < s) smem[threadIdx.x] += smem[threadIdx.x + s];
        __syncthreads();
    }
    if (threadIdx.x == 0) pooled[b * C_ + c] = smem[0];
}

// ---------------------------------------------------------------------------
// Pass 4: bottleneck MLP via V_WMMA_F32_16X16X4_F32 (fp32 WMMA, wave32).
//   v  = pooled(32x512) @ w2^T(512x32) + b2   -> LN(32) -> ReLU
//   v2 = v(32x32)       @ w3^T(32x512) + b3
// Single workgroup, 128 threads = 4 waves, full EXEC around all WMMAs.
// A-operand 16x4 f32 layout: lane m = lane[3:0], VGPR v holds K = 2*lane[4]+v.
// B-operand 4x16 f32 mirrors with lane n = lane[3:0].
// D 16x16 f32: element (m = r + 8*lane[4], n = lane[3:0]) in VGPR r.
// ---------------------------------------------------------------------------
__global__ __launch_bounds__(128) void gc_bottleneck(const float* __restrict__ pooled,
                                                     const float* __restrict__ w2,
                                                     const float* __restrict__ b2,
                                                     const float* __restrict__ ln_g,
                                                     const float* __restrict__ ln_b,
                                                     const float* __restrict__ w3,
                                                     const float* __restrict__ b3,
                                                     float* __restrict__ v2out) {
    __shared__ float vbuf[C16_ * C16_];          // 32x32 bottleneck activations
    const int lane = threadIdx.x & 31;
    const int wave = threadIdx.x >> 5;
    const int l15  = lane & 15;
    const int hi   = lane >> 4;
    const int koff = hi * 2;                     // K pair selected by lane[4]

    // ---- GEMM1: M=32, N=32, K=512 ; wave -> one 16x16 output tile (2x2) ----
    {
        const int mt = wave >> 1, nt = wave & 1;
        const int mm = mt * 16 + l15;            // batch row (A)
        const int nn = nt * 16 + l15;            // C16 col  (B)
        v8f acc = {};
        for (int k0 = 0; k0 < C_; k0 += 4) {
            v2f a = *(const v2f*)(pooled + mm * C_ + k0 + koff);
            v2f b = *(const v2f*)(w2 + nn * C_ + k0 + koff);   // w2^T[k][nn] = w2[nn][k]
            acc = __builtin_amdgcn_wmma_f32_16x16x4_f32(
                false, a, false, b, (short)0, acc, false, false);
        }
        const float bias = b2[nn];
        #pragma unroll
        for (int r = 0; r < 8; ++r) {
            const int m = mt * 16 + hi * 8 + r;
            vbuf[m * C16_ + nn] = acc[r] + bias;
        }
    }
    __syncthreads();

    // ---- LayerNorm over C16 + ReLU (one thread per batch row) ----
    if (threadIdx.x < B_) {
        float* row = vbuf + threadIdx.x * C16_;
        float mu = 0.0f;
        #pragma unroll
        for (int j = 0; j < C16_; ++j) mu += row[j];
        mu *= (1.0f / C16_);
        float var = 0.0f;
        #pragma unroll
        for (int j = 0; j < C16_; ++j) { float d = row[j] - mu; var += d * d; }
        var *= (1.0f / C16_);
        const float inv = rsqrtf(var + LN_EPS_);
        #pragma unroll
        for (int j = 0; j < C16_; ++j) {
            float t = (row[j] - mu) * inv * ln_g[j] + ln_b[j];
            row[j] = t > 0.0f ? t : 0.0f;
        }
    }
    __syncthreads();

    // ---- GEMM2: M=32, N=512, K=32 ; 64 tiles round-robined over 4 waves ----
    for (int t = wave; t < 64; t += 4) {
        const int mt = t & 1;
        const int nt = t >> 1;
        const int mm = mt * 16 + l15;
        const int nn = nt * 16 + l15;            // output channel (B from w3)
        v8f acc = {};
        #pragma unroll
        for (int k0 = 0; k0 < C16_; k0 += 4) {
            v2f a = *(const v2f*)&vbuf[mm * C16_ + k0 + koff];
            v2f b = *(const v2f*)(w3 + nn * C16_ + k0 + koff); // w3^T[k][nn] = w3[nn][k]
            acc = __builtin_amdgcn_wmma_f32_16x16x4_f32(
                false, a, false, b, (short)0, acc, false, false);
        }
        const float bias = b3[nn];
        #pragma unroll
        for (int r = 0; r < 8; ++r) {
            const int m = mt * 16 + hi * 8 + r;
            v2out[m * C_ + nn] = acc[r] + bias;
        }
    }
}

// ---------------------------------------------------------------------------
// Pass 5: out = x + v2[b,c]  (float4 per thread; v2 is L2-resident)
// grid: (B*C*HW/4)/256 = 65536 blocks.
// ---------------------------------------------------------------------------
__global__ __launch_bounds__(256) void gc_addbcast(const float* __restrict__ x,
                                                   const float* __restrict__ v2,
                                                   float* __restrict__ out) {
    const int i = blockIdx.x * 256 + threadIdx.x;    // float4 index, < 2^24
    const int bc = i >> 10;                          // HW/4 = 1024 float4 per (b,c)
    const float add = v2[bc];
    float4 xv = ((const float4*)x)[i];
    xv.x += add; xv.y += add; xv.z += add; xv.w += add;
    ((float4*)out)[i] = xv;
}

// ---------------------------------------------------------------------------
extern "C" void kernel_launch(void* const* d_in, const int* in_sizes, int n_in,
                              void* d_out, int out_size, void* d_ws, size_t ws_size,
                              hipStream_t stream) {
    const float* x    = (const float*)d_in[0];
    const float* w1   = (const float*)d_in[1];
    const float* b1   = (const float*)d_in[2];
    const float* w2   = (const float*)d_in[3];
    const float* b2   = (const float*)d_in[4];
    const float* ln_g = (const float*)d_in[5];
    const float* ln_b = (const float*)d_in[6];
    const float* w3   = (const float*)d_in[7];
    const float* b3   = (const float*)d_in[8];
    float* out = (float*)d_out;

    // workspace layout (fp32): kbuf[B*HW] | pooled[B*C] | v2[B*C]  (~640 KB)
    float* kbuf   = (float*)d_ws;
    float* pooled = kbuf + B_ * HW_;
    float* v2     = pooled + B_ * C_;

    gc_logits<<<B_ * (HW_ / 256), 256, 0, stream>>>(x, w1, b1, kbuf);
    gc_softmax<<<B_, 256, 0, stream>>>(kbuf);
    gc_pool<<<B_ * C_, 128, 0, stream>>>(x, kbuf, pooled);
    gc_bottleneck<<<1, 128, 0, stream>>>(pooled, w2, b2, ln_g, ln_b, w3, b3, v2);
    gc_addbcast<<<(B_ * C_ * HW_ / 4) / 256, 256, 0, stream>>>(x, v2, out);
}